// QuantumEnhancedTransformerBlock_9053791060383
// MI455X (gfx1250) — compile-verified
//
#include <hip/hip_runtime.h>
#include <hip/hip_bf16.h>
#include <math.h>

typedef __attribute__((ext_vector_type(2))) float v2f;
typedef __attribute__((ext_vector_type(8))) float v8f;

// ---------------------------------------------------------------------------
// Kernel: E = exp(i * entanglement)  ->  cos / sin planes (128x128 each)
// ---------------------------------------------------------------------------
__global__ void prep_e_kernel(const float* __restrict__ ent,
                              float* __restrict__ ec, float* __restrict__ es) {
    int i = blockIdx.x * 256 + threadIdx.x;
    if (i < 128 * 128) {
        float a = ent[i];
        ec[i] = cosf(a);
        es[i] = sinf(a);
    }
}

// ---------------------------------------------------------------------------
// Kernel: T = I (complex identity, separate re/im planes)
// ---------------------------------------------------------------------------
__global__ void init_t_kernel(float* __restrict__ re, float* __restrict__ im) {
    int i = blockIdx.x * 256 + threadIdx.x;
    if (i < 128 * 128) {
        int r = i >> 7, c = i & 127;
        re[i] = (r == c) ? 1.0f : 0.0f;
        im[i] = 0.0f;
    }
}

// ---------------------------------------------------------------------------
// Kernel: apply the 7 non-identity qubit rotations of one layer, in place.
// T = T @ R_q  only touches column pairs (i, i|mask); pairs are disjoint so
// each stage is embarrassingly parallel; stages separated by __syncthreads.
// Single block (work is tiny: 7 * 8192 pair updates).
// ---------------------------------------------------------------------------
__global__ __launch_bounds__(256) void rotate_kernel(float* __restrict__ re,
                                                     float* __restrict__ im,
                                                     const float* __restrict__ rp,
                                                     int layer) {
    const int tid = threadIdx.x;
    for (int q = 121; q < 128; ++q) {
        const int mask = 1 << (127 - q);          // 64, 32, ..., 1
        const float* p3 = rp + ((size_t)layer * 128 + q) * 3;
        float phi = p3[0], th = p3[1], om = p3[2];
        float ct = cosf(0.5f * th), st = sinf(0.5f * th);
        float apo = 0.5f * (phi + om), amo = 0.5f * (phi - om);
        float capo = cosf(apo), sapo = sinf(apo);
        float camo = cosf(amo), samo = sinf(amo);
        // m00 = ct*e^{-i*apo}, m01 = -st*e^{+i*amo}, m10 = st*e^{-i*amo}, m11 = ct*e^{+i*apo}
        float m00r = ct * capo,  m00i = -ct * sapo;
        float m01r = -st * camo, m01i = -st * samo;
        float m10r = st * camo,  m10i = -st * samo;
        float m11r = ct * capo,  m11i = ct * sapo;
        __syncthreads();
        for (int it = tid; it < 128 * 64; it += 256) {
            int r = it >> 6, p = it & 63;
            int i0 = ((p & ~(mask - 1)) << 1) | (p & (mask - 1));
            int i1 = i0 | mask;
            int a0 = r * 128 + i0, a1 = r * 128 + i1;
            float aR = re[a0], aI = im[a0];
            float bR = re[a1], bI = im[a1];
            // col i0 <- a*m00 + b*m10 ; col i1 <- a*m01 + b*m11   (complex)
            re[a0] = aR * m00r - aI * m00i + bR * m10r - bI * m10i;
            im[a0] = aR * m00i + aI * m00r + bR * m10i + bI * m10r;
            re[a1] = aR * m01r - aI * m01i + bR * m11r - bI * m11i;
            im[a1] = aR * m01i + aI * m01r + bR * m11i + bI * m11r;
        }
    }
}

// ---------------------------------------------------------------------------
// Kernel: out = in @ E  (complex 128x128x128). One thread per output element.
// On the final layer, write the packed real matrix Tr[128][256] = [Re | Im].
// ---------------------------------------------------------------------------
__global__ __launch_bounds__(256) void matmul_e_kernel(const float* __restrict__ inRe,
                                                       const float* __restrict__ inIm,
                                                       const float* __restrict__ ec,
                                                       const float* __restrict__ es,
                                                       float* __restrict__ outRe,
                                                       float* __restrict__ outIm,
                                                       float* __restrict__ packed) {
    int g = blockIdx.x * 256 + threadIdx.x;   // 0 .. 16383
    int r = g >> 7, c = g & 127;
    float sr = 0.0f, si = 0.0f;
    for (int k = 0; k < 128; ++k) {
        float tr = inRe[r * 128 + k], ti = inIm[r * 128 + k];
        float C = ec[k * 128 + c],    S = es[k * 128 + c];
        sr = fmaf(tr, C, fmaf(-ti, S, sr));
        si = fmaf(tr, S, fmaf(ti, C, si));
    }
    if (packed) {
        packed[r * 256 + c]       = sr;
        packed[r * 256 + 128 + c] = si;
    } else {
        outRe[g] = sr;
        outIm[g] = si;
    }
}

// ---------------------------------------------------------------------------
// Main kernel: Y = sqrt(x+1e-6) @ Tr  (65536x128 @ 128x256, fp32 WMMA),
// p = Yre^2 + Yim^2, row-normalize, replicate 4x into out.
// Block: 256 threads = 8 waves, 16 rows. Wave w owns output cols [32w,32w+32).
// All global load/store in the epilogue is b128-vectorized.
// ---------------------------------------------------------------------------
__global__ __launch_bounds__(256) void qmm_kernel(const float* __restrict__ x,
                                                  const float* __restrict__ Tr,
                                                  float* __restrict__ out) {
    __shared__ float Sld[16][128];   // sqrt tile (8 KB)
    __shared__ float Y[16][256];     // Yre | Yim  (16 KB)

    const int tid = threadIdx.x;
    const int m0  = blockIdx.x << 4;

    // stage S = sqrt(x + 1e-6) using float4 loads/stores (2 per thread)
    const float4* x4 = (const float4*)x;
    for (int it = tid; it < 16 * 32; it += 256) {
        int r = it >> 5, c4 = it & 31;
        float4 v = x4[(size_t)(m0 + r) * 32 + c4];
        float4 s;
        s.x = sqrtf(v.x + 1e-6f);
        s.y = sqrtf(v.y + 1e-6f);
        s.z = sqrtf(v.z + 1e-6f);
        s.w = sqrtf(v.w + 1e-6f);
        ((float4*)&Sld[r][0])[c4] = s;
    }
    __syncthreads();

    const int lane = tid & 31;
    const int wave = tid >> 5;
    const int lm   = lane & 15;     // M / N within tile
    const int hg   = lane >> 4;     // half-wave group -> K offset of 2
    const int n0   = wave << 5;     // 32 columns per wave

    v8f acc0 = {0,0,0,0,0,0,0,0};
    v8f acc1 = {0,0,0,0,0,0,0,0};

    for (int k = 0; k < 128; k += 4) {
        // A frag: element (m=lm, k = k + 2*hg + {0,1})
        v2f a;
        a.x = Sld[lm][k + 2 * hg];
        a.y = Sld[lm][k + 2 * hg + 1];
        // B frags: element (k = k + 2*hg + {0,1}, n = n0 + lm / +16)
        const float* t0 = Tr + (size_t)(k + 2 * hg) * 256;
        v2f b0, b1;
        b0.x = t0[n0 + lm];        b0.y = t0[256 + n0 + lm];
        b1.x = t0[n0 + 16 + lm];   b1.y = t0[256 + n0 + 16 + lm];

        acc0 = __builtin_amdgcn_wmma_f32_16x16x4_f32(
            false, a, false, b0, (short)0, acc0, false, false);
        acc1 = __builtin_amdgcn_wmma_f32_16x16x4_f32(
            false, a, false, b1, (short)0, acc1, false, false);
    }

    // C/D layout: row = v + 8*hg, col = lm
#pragma unroll
    for (int v = 0; v < 8; ++v) {
        Y[v + 8 * hg][n0 + lm]      = acc0[v];
        Y[v + 8 * hg][n0 + 16 + lm] = acc1[v];
    }
    __syncthreads();

    // p = re^2 + im^2, normalize per row, write 4 head-copies (b128 stores).
    const int row = tid >> 4;       // 16 threads per row
    const int j   = tid & 15;       // thread owns cols [4j,4j+4) + 64*cc
    float4 pv[2];
    float sum = 0.0f;
#pragma unroll
    for (int cc = 0; cc < 2; ++cc) {
        int c = (j << 2) + (cc << 6);
        float4 yr = *(const float4*)&Y[row][c];
        float4 yi = *(const float4*)&Y[row][c + 128];
        float4 p;
        p.x = yr.x * yr.x + yi.x * yi.x;
        p.y = yr.y * yr.y + yi.y * yi.y;
        p.z = yr.z * yr.z + yi.z * yi.z;
        p.w = yr.w * yr.w + yi.w * yi.w;
        pv[cc] = p;
        sum += p.x + p.y + p.z + p.w;
    }
    sum += __shfl_xor(sum, 1, 32);
    sum += __shfl_xor(sum, 2, 32);
    sum += __shfl_xor(sum, 4, 32);
    sum += __shfl_xor(sum, 8, 32);
    const float inv = 1.0f / sum;

    float4* orow = (float4*)(out + (size_t)(m0 + row) * 512);
#pragma unroll
    for (int cc = 0; cc < 2; ++cc) {
        float4 v;
        v.x = pv[cc].x * inv;
        v.y = pv[cc].y * inv;
        v.z = pv[cc].z * inv;
        v.w = pv[cc].w * inv;
        int c4 = j + (cc << 4);            // float4 index within 128-col head
        orow[c4]      = v;                 // head 0
        orow[c4 + 32] = v;                 // head 1
        orow[c4 + 64] = v;                 // head 2
        orow[c4 + 96] = v;                 // head 3
    }
}

// ---------------------------------------------------------------------------
// Host launcher
// ---------------------------------------------------------------------------
extern "C" void kernel_launch(void* const* d_in, const int* in_sizes, int n_in,
                              void* d_out, int out_size, void* d_ws, size_t ws_size,
                              hipStream_t stream) {
    const float* x   = (const float*)d_in[0];   // (16, 4096, 128)
    const float* rp  = (const float*)d_in[1];   // (5, 128, 3)
    const float* ent = (const float*)d_in[2];   // (128, 128)
    float* out = (float*)d_out;

    char* ws = (char*)d_ws;
    float* Tr  = (float*)(ws);             // 128*256 packed [Re|Im]   (128 KB)
    float* Ec  = (float*)(ws + 131072);    // cos(ent)                 ( 64 KB)
    float* Es  = (float*)(ws + 196608);    // sin(ent)                 ( 64 KB)
    float* Are = (float*)(ws + 262144);    // ping complex planes      (128 KB)
    float* Aim = Are + 16384;
    float* Bre = (float*)(ws + 393216);    // pong complex planes      (128 KB)
    float* Bim = Bre + 16384;

    prep_e_kernel<<<64, 256, 0, stream>>>(ent, Ec, Es);
    init_t_kernel<<<64, 256, 0, stream>>>(Are, Aim);

    float* curRe = Are; float* curIm = Aim;
    float* nxtRe = Bre; float* nxtIm = Bim;
    for (int layer = 0; layer < 5; ++layer) {
        rotate_kernel<<<1, 256, 0, stream>>>(curRe, curIm, rp, layer);
        const bool last = (layer == 4);
        matmul_e_kernel<<<64, 256, 0, stream>>>(curRe, curIm, Ec, Es,
                                                nxtRe, nxtIm,
                                                last ? Tr : (float*)nullptr);
        float* t;
        t = curRe; curRe = nxtRe; nxtRe = t;
        t = curIm; curIm = nxtIm; nxtIm = t;
    }

    const int rows = in_sizes[0] / 128;        // 65536
    qmm_kernel<<<rows / 16, 256, 0, stream>>>(x, Tr, out);
}